// FixedDb4DWT1D_71390946394514
// MI455X (gfx1250) — compile-verified
//
#include <hip/hip_runtime.h>

// MI455X (gfx1250) fused 3-level db4 DWT + bilinear upsample + concat.
// Memory-bound: ~384MB min HBM traffic -> single fused kernel, all
// intermediates in LDS, convs done on the matrix pipe via
// v_wmma_f32_16x16x4_f32 (D[16x16] = Mfilter[16x40] @ Xwin[40x16], 10 chunks).
// x staged global->LDS with GLOBAL_LOAD_ASYNC_TO_LDS_B128 when available.

typedef __attribute__((ext_vector_type(2))) float v2f;
typedef __attribute__((ext_vector_type(8))) float v8f;
typedef __attribute__((ext_vector_type(4))) int   v4i;

#if defined(__has_builtin)
#if __has_builtin(__builtin_amdgcn_global_load_async_to_lds_b128) && \
    __has_builtin(__builtin_amdgcn_s_wait_asynccnt)
#define DWT_ASYNC_LDS 1
#endif
#endif
#ifndef DWT_ASYNC_LDS
#define DWT_ASYNC_LDS 0
#endif

__device__ __constant__ float c_dec_lo[8] = {
    -0.010597401784997278f, 0.032883011666982945f,
     0.030841381835986965f, -0.18703481171888114f,
    -0.02798376941698385f,  0.6308807679295904f,
     0.7148465705525415f,   0.23037781330885523f};
__device__ __constant__ float c_dec_hi[8] = {
    -0.23037781330885523f,  0.7148465705525415f,
    -0.6308807679295904f,  -0.02798376941698385f,
     0.18703481171888114f,  0.030841381835986965f,
     0.032883011666982945f, -0.010597401784997278f};

// Problem geometry
constexpr int L_IN = 16384;   // input length
constexpr int L1   = 8192;    // level-1 length (== output length)
constexpr int L2N  = 4096;    // level-2 length
constexpr int L3N  = 2048;    // level-3 length
constexpr int C    = 64;      // channels
constexpr int TT   = 64;      // output time positions per workgroup

// Staged LDS window sizes (rows) — exact halo coverage verified:
//   s in [t0-16, t0+96)  covers consumed [t0-13, t0+76]
//   u in [t0/2-8, t0/2+40) covers consumed [t0/2-5, t0/2+36]
//   v in [t0/4-4, t0/4+28) covers consumed [t0/4-1, t0/4+16]
//   x in [2t0-35, 2t0+173) covers consumed [2t0-29, 2t0+156]
// Padded (never-consumed) tile lanes may read a few rows past their own
// sub-buffer; all such reads stay inside the total LDS allocation
// (worst case: Xs + 231 rows = 59KB < 120KB total), so no clamping needed.
constexpr int S_CNT = 112;
constexpr int U_CNT = 48;
constexpr int V_CNT = 32;
constexpr int X_CNT = 208;

static __device__ inline int clampi(int v, int lo, int hi) {
  return v < lo ? lo : (v > hi ? hi : v);
}

__global__ __launch_bounds__(256) void dwt_fused(const float* __restrict__ x,
                                                 float* __restrict__ out) {
  const int b    = blockIdx.y;
  const int t0   = blockIdx.x * TT;
  const int tid  = threadIdx.x;
  const int lane = tid & 31;          // wave32
  const int wave = tid >> 5;          // 8 waves / block
  const int half = lane >> 4;         // lane half selects K-pair / M-half
  const int n16  = lane & 15;         // column (channel-within-block / M row)

  const int s_b = t0 - 16;
  const int u_b = (t0 >> 1) - 8;
  const int v_b = (t0 >> 2) - 4;
  const int x_b = 2 * t0 - 35;        // == 2*s_b - 3

  extern __shared__ float smem[];
  float* Xs  = smem;                  // X_CNT * C
  float* A1s = Xs  + X_CNT * C;       // S_CNT * C
  float* A2s = A1s + S_CNT * C;       // U_CNT * C
  float* D2s = A2s + U_CNT * C;       // U_CNT * C
  float* A3s = D2s + U_CNT * C;       // V_CNT * C
  float* D3s = A3s + V_CNT * C;       // V_CNT * C

  // ---- Per-lane A-operand constants: M[m][r] = h[r - 2m], r in [0,40) ----
  // A 16x4 f32 layout: lanes 0-15 hold (K=0,K=1), lanes 16-31 hold (K=2,K=3).
  v2f aL[10], aH[10];
  {
    const int m2   = 2 * n16;
    const int koff = half * 2;
#pragma unroll
    for (int c = 0; c < 10; ++c) {
      const int r0 = 4 * c + koff;
      const int d0 = r0 - m2;
      const int d1 = r0 + 1 - m2;
      aL[c].x = (d0 >= 0 && d0 < 8) ? c_dec_lo[d0] : 0.0f;
      aL[c].y = (d1 >= 0 && d1 < 8) ? c_dec_lo[d1] : 0.0f;
      aH[c].x = (d0 >= 0 && d0 < 8) ? c_dec_hi[d0] : 0.0f;
      aH[c].y = (d1 >= 0 && d1 < 8) ? c_dec_hi[d1] : 0.0f;
    }
  }

  // ---- Stage 0: stage x window into LDS (zero pad outside [0, L_IN)) ----
  {
    const float* xb = x + (size_t)b * L_IN * C;
    for (int e = tid; e < X_CNT * (C / 4); e += 256) {
      const int row = e >> 4;       // C/4 == 16 float4 per row
      const int c4  = e & 15;
      const int n   = x_b + row;
#if DWT_ASYNC_LDS
      if (n >= 0 && n < L_IN) {
        // GLOBAL_LOAD_ASYNC_TO_LDS_B128: memory -> LDS, no VGPR round-trip.
        __builtin_amdgcn_global_load_async_to_lds_b128(
            (__attribute__((address_space(1))) v4i*)
                (xb + (size_t)n * C + c4 * 4),
            (__attribute__((address_space(3))) v4i*)
                (Xs + row * C + c4 * 4),
            0, 0);
      } else {
        ((float4*)Xs)[row * 16 + c4] = make_float4(0.f, 0.f, 0.f, 0.f);
      }
#else
      float4 v = make_float4(0.f, 0.f, 0.f, 0.f);
      if (n >= 0 && n < L_IN) v = ((const float4*)(xb + (size_t)n * C))[c4];
      ((float4*)Xs)[row * 16 + c4] = v;
#endif
    }
#if DWT_ASYNC_LDS
    __builtin_amdgcn_s_wait_asynccnt(0);
#endif
  }
  __syncthreads();

  // Dual-filter 16x16 conv tile on the matrix pipe.
  // D[m][n] = sum_r M[m][r] * src[base0 + r][cb + n], r in [0,40).
  // B 4x16 layout assumed: VGPR j, lane l -> K = 4c + 2*(l>=16) + j, N = l%16.
  // One base pointer per tile; per-chunk row offsets become ds_load
  // immediate offsets (4c*256B <= 9.5KB, fits the 16-bit offset field).
  auto conv_tile = [&](const float* src, int base0, int cb,
                       v8f& dLo, v8f& dHi) {
    const float* p = src + (base0 + 2 * half) * C + cb + n16;
    v8f accL = {0, 0, 0, 0, 0, 0, 0, 0};
    v8f accH = {0, 0, 0, 0, 0, 0, 0, 0};
#pragma unroll
    for (int c = 0; c < 10; ++c) {
      v2f bv;
      bv.x = p[(4 * c + 0) * C];
      bv.y = p[(4 * c + 1) * C];
      accL = __builtin_amdgcn_wmma_f32_16x16x4_f32(
          false, aL[c], false, bv, (short)0, accL, false, false);
      accH = __builtin_amdgcn_wmma_f32_16x16x4_f32(
          false, aH[c], false, bv, (short)0, accH, false, false);
    }
    dLo = accL;
    dHi = accH;
  };

  // ---- Level 1: x -> A1 (LDS, zero outside [0,L1)), D1 -> out[ch 0:64] ----
  for (int job = wave; job < 7 * 4; job += 8) {  // 7 time tiles x 4 ch blocks
    const int ts = (job >> 2) * 16;
    const int cb = (job & 3) * 16;
    v8f aAcc, dAcc;
    conv_tile(Xs, 2 * ts, cb, aAcc, dAcc);
#pragma unroll
    for (int r = 0; r < 8; ++r) {
      const int sl = ts + r + half * 8;   // D layout: M = r + 8*half
      const int sg = s_b + sl;
      float av = aAcc[r];
      if (sg < 0 || sg >= L1) av = 0.0f;  // level-2 zero padding
      A1s[sl * C + cb + n16] = av;
      if (sg >= t0 && sg < t0 + TT)       // D1 (identity resize) direct out
        out[((size_t)(b * L1 + sg)) * 256 + cb + n16] = dAcc[r];
    }
  }
  __syncthreads();

  // ---- Level 2: A1 -> A2 (zeroed outside [0,L2N)), D2 (LDS) ----
  for (int job = wave; job < 3 * 4; job += 8) {
    const int tu = (job >> 2) * 16;
    const int cb = (job & 3) * 16;
    v8f aAcc, dAcc;
    conv_tile(A1s, 2 * tu - 3, cb, aAcc, dAcc);
#pragma unroll
    for (int r = 0; r < 8; ++r) {
      const int ul = tu + r + half * 8;
      const int ug = u_b + ul;
      float av = aAcc[r];
      if (ug < 0 || ug >= L2N) av = 0.0f;
      A2s[ul * C + cb + n16] = av;
      D2s[ul * C + cb + n16] = dAcc[r];
    }
  }
  __syncthreads();

  // ---- Level 3: A2 -> A3, D3 (LDS) ----
  for (int job = wave; job < 2 * 4; job += 8) {
    const int tv = (job >> 2) * 16;
    const int cb = (job & 3) * 16;
    v8f aAcc, dAcc;
    conv_tile(A2s, 2 * tv - 3, cb, aAcc, dAcc);
#pragma unroll
    for (int r = 0; r < 8; ++r) {
      const int vl = tv + r + half * 8;
      A3s[vl * C + cb + n16] = aAcc[r];
      D3s[vl * C + cb + n16] = dAcc[r];
    }
  }
  __syncthreads();

  // ---- Upsample (bilinear, half-pixel, edge clamp) + write out ----
  // part 0: D2 x2 -> ch[64:128]; part 1: D3 x4 -> ch[128:192];
  // part 2: A3 x4 -> ch[192:256]
  for (int e = tid; e < TT * 48; e += 256) {   // 48 = 3 parts * 16 float4
    const int q    = e % 48;
    const int toff = e / 48;
    const int part = q / 16;
    const int c4   = q % 16;
    const int t    = t0 + toff;
    int i0, num, basei;
    float w;
    if (part == 0) {                 // pos = t/2 - 0.25 = (2t-1)/4
      const int p = 2 * t - 1;
      i0 = p >> 2; w = (float)(p & 3) * 0.25f; num = L2N; basei = u_b;
    } else {                         // pos = t/4 - 0.375 = (2t-3)/8
      const int p = 2 * t - 3;
      i0 = p >> 3; w = (float)(p & 7) * 0.125f; num = L3N; basei = v_b;
    }
    const int g0 = clampi(i0, 0, num - 1);
    const int g1 = clampi(i0 + 1, 0, num - 1);
    const float* sp = (part == 0) ? D2s : ((part == 1) ? D3s : A3s);
    const float4 s0 = ((const float4*)sp)[(g0 - basei) * 16 + c4];
    const float4 s1 = ((const float4*)sp)[(g1 - basei) * 16 + c4];
    float4 r;
    r.x = s0.x * (1.0f - w) + s1.x * w;
    r.y = s0.y * (1.0f - w) + s1.y * w;
    r.z = s0.z * (1.0f - w) + s1.z * w;
    r.w = s0.w * (1.0f - w) + s1.w * w;
    ((float4*)(out + ((size_t)(b * L1 + t)) * 256 + 64 + part * 64))[c4] = r;
  }
}

extern "C" void kernel_launch(void* const* d_in, const int* in_sizes, int n_in,
                              void* d_out, int out_size, void* d_ws,
                              size_t ws_size, hipStream_t stream) {
  (void)in_sizes; (void)n_in; (void)out_size; (void)d_ws; (void)ws_size;
  const float* x = (const float*)d_in[0];
  float* out = (float*)d_out;
  const dim3 grid(L1 / TT, 32);      // 128 time tiles x 32 batches
  const dim3 block(256);             // 8 wave32 waves
  const size_t shmem =
      (size_t)(X_CNT + S_CNT + 2 * U_CNT + 2 * V_CNT) * C * sizeof(float);
  hipLaunchKernelGGL(dwt_fused, grid, block, shmem, stream, x, out);
}